// BrainNN_74345883894230
// MI455X (gfx1250) — compile-verified
//
#include <hip/hip_runtime.h>
#include <hip/hip_bf16.h>

// ---------------------------------------------------------------------------
// BrainNN on MI455X (gfx1250, wave32, WMMA).
// GEMMs use V_WMMA_F32_16X16X4_F32 (exact f32). The relu-inside-sum pairwise
// message term is pure VALU with LDS-cached W tiles.
// ---------------------------------------------------------------------------

#define B_   8
#define N_   256
#define DIN_ 256
#define HID_ 128
#define MSG_ 256
#define RO_  128
#define L_   2
#define C_   2

typedef __attribute__((ext_vector_type(2))) float v2f;
typedef __attribute__((ext_vector_type(8))) float v8f;

// ---------------------------------------------------------------------------
// WMMA f32 GEMM:  out[m, o] = epilogue( sum_k A[m,k] * Wt[o,k] )
//   A: [M, K] row-major (lda), Wt: [OUT, K] row-major (ldw) -> B[k,o]=Wt[o,k]
// Block: 128 threads = 4 waves; wave w computes rows [m0+16w, +16), cols 16.
// Grid: (M/64, OUT/16).
// EPI: 0 = none, 1 = bias+relu, 2 = bias, 3 = bias*bscale + relu + residual
// ---------------------------------------------------------------------------
template <int EPI>
__global__ __launch_bounds__(128)
void wmma_gemm_f32(const float* __restrict__ A, int lda,
                   const float* __restrict__ Wt, int ldw,
                   const float* __restrict__ bias, float bscale,
                   const float* __restrict__ resid,
                   float* __restrict__ out, int ldo, int K)
{
    const int lane    = threadIdx.x & 31;
    const int wave    = threadIdx.x >> 5;
    const int laneLow = lane & 15;
    const int hi      = lane >> 4;            // 0: K pair (0,1), 1: K pair (2,3)

    const int m0 = blockIdx.x * 64 + wave * 16;
    const int n0 = blockIdx.y * 16;

    // A fragment source row (M = laneLow), B fragment source row (N = laneLow)
    const float* Arow = A  + (size_t)(m0 + laneLow) * lda + hi * 2;
    const float* Brow = Wt + (size_t)(n0 + laneLow) * ldw + hi * 2;

    v8f acc = {};
    for (int k = 0; k < K; k += 4) {
        v2f av, bv;
        av.x = Arow[k];     av.y = Arow[k + 1];
        bv.x = Brow[k];     bv.y = Brow[k + 1];
        // D = A(16x4) * B(4x16) + C   -> v_wmma_f32_16x16x4_f32
        acc = __builtin_amdgcn_wmma_f32_16x16x4_f32(
            false, av, false, bv, (short)0, acc, false, false);
    }

    // C/D layout: VGPR r -> row m0 + r + 8*hi, col n0 + laneLow
    const int col  = n0 + laneLow;
    const int rowb = m0 + 8 * hi;
    float bcol = 0.0f;
    if (EPI != 0) bcol = bias[col] * bscale;

#pragma unroll
    for (int r = 0; r < 8; ++r) {
        float v = acc[r];
        if (EPI != 0) v += bcol;
        if (EPI == 1 || EPI == 3) v = fmaxf(v, 0.0f);
        if (EPI == 3) v += resid[(size_t)(rowb + r) * ldo + col];
        out[(size_t)(rowb + r) * ldo + col] = v;
    }
}

// ---------------------------------------------------------------------------
// Pairwise message kernel (VALU-bound part):
//   s[b,n,m] = sum_j relu( a[b,n,m] + bArr[b,j,m] + W[b,n,j]*ww[m] + bias1[m] )
// Block: 256 threads (m dimension), handles 16 consecutive n rows of one batch.
// W tile (16 x 256 = 16 KB) cached in LDS; bArr[b,j,:] load is reused across
// all 16 rows -> 4 VALU ops per (n,j) element.
// Grid: (N/16, B)
// ---------------------------------------------------------------------------
__global__ __launch_bounds__(256)
void pairwise_msg(const float* __restrict__ aArr,
                  const float* __restrict__ bArr,
                  const float* __restrict__ W,
                  const float* __restrict__ w1l,     // msg_w1[l], row stride 257
                  const float* __restrict__ bias1,   // msg_b1[l]
                  float* __restrict__ s)
{
    __shared__ float Wtile[16 * N_];   // 16 KB

    const int b  = blockIdx.y;
    const int n0 = blockIdx.x * 16;
    const int m  = threadIdx.x;        // 0..255

    // Cooperative load of W[b, n0..n0+15, :] into LDS (coalesced).
    const float* Wb = W + (size_t)b * N_ * N_;
#pragma unroll
    for (int i = 0; i < 16; ++i)
        Wtile[i * N_ + m] = Wb[(size_t)(n0 + i) * N_ + m];
    __syncthreads();

    const float wwm = w1l[(size_t)m * (2 * HID_ + 1) + 2 * HID_]; // ww[m]
    const float bs  = bias1[m];

    float c[16];
#pragma unroll
    for (int i = 0; i < 16; ++i)
        c[i] = aArr[((size_t)(b * N_ + n0 + i)) * MSG_ + m] + bs;

    float acc[16];
#pragma unroll
    for (int i = 0; i < 16; ++i) acc[i] = 0.0f;

    for (int j = 0; j < N_; ++j) {
        const float bb = bArr[((size_t)(b * N_ + j)) * MSG_ + m];
#pragma unroll
        for (int i = 0; i < 16; ++i) {
            float v = fmaf(Wtile[i * N_ + j], wwm, c[i] + bb);
            acc[i] += fmaxf(v, 0.0f);
        }
    }

#pragma unroll
    for (int i = 0; i < 16; ++i)
        s[((size_t)(b * N_ + n0 + i)) * MSG_ + m] = acc[i];
}

// ---------------------------------------------------------------------------
// Readout: z' = sum_n h ; r = relu(z' W1^T + b1) W2^T + b2 ; z = r + z'
//          out = z cls_w^T + cls_b.   Grid: (B), block: 128 threads.
// ---------------------------------------------------------------------------
__global__ __launch_bounds__(128)
void readout(const float* __restrict__ h,
             const float* __restrict__ ro_w1, const float* __restrict__ ro_b1,
             const float* __restrict__ ro_w2, const float* __restrict__ ro_b2,
             const float* __restrict__ cls_w, const float* __restrict__ cls_b,
             float* __restrict__ out)
{
    __shared__ float zp[RO_], u[RO_], z[RO_];
    const int b = blockIdx.x;
    const int t = threadIdx.x;

    float s = 0.0f;
    for (int n = 0; n < N_; ++n)
        s += h[((size_t)(b * N_ + n)) * HID_ + t];
    zp[t] = s;
    __syncthreads();

    float a1 = ro_b1[t];
    for (int k = 0; k < HID_; ++k)
        a1 = fmaf(zp[k], ro_w1[(size_t)t * HID_ + k], a1);
    u[t] = fmaxf(a1, 0.0f);
    __syncthreads();

    float a2 = ro_b2[t];
    for (int k = 0; k < RO_; ++k)
        a2 = fmaf(u[k], ro_w2[(size_t)t * RO_ + k], a2);
    z[t] = a2 + zp[t];
    __syncthreads();

    if (t < C_) {
        float o = cls_b[t];
        for (int k = 0; k < RO_; ++k)
            o = fmaf(z[k], cls_w[(size_t)t * RO_ + k], o);
        out[b * C_ + t] = o;
    }
}

// ---------------------------------------------------------------------------
extern "C" void kernel_launch(void* const* d_in, const int* in_sizes, int n_in,
                              void* d_out, int out_size, void* d_ws, size_t ws_size,
                              hipStream_t stream)
{
    (void)in_sizes; (void)n_in; (void)out_size; (void)ws_size;

    const float* W      = (const float*)d_in[0];
    const float* X      = (const float*)d_in[1];
    const float* enc_w1 = (const float*)d_in[2];
    const float* enc_b1 = (const float*)d_in[3];
    const float* enc_w2 = (const float*)d_in[4];
    const float* enc_b2 = (const float*)d_in[5];
    const float* msg_w1 = (const float*)d_in[6];
    const float* msg_b1 = (const float*)d_in[7];
    const float* msg_w2 = (const float*)d_in[8];
    const float* msg_b2 = (const float*)d_in[9];
    const float* ro_w1  = (const float*)d_in[10];
    const float* ro_b1  = (const float*)d_in[11];
    const float* ro_w2  = (const float*)d_in[12];
    const float* ro_b2  = (const float*)d_in[13];
    const float* cls_w  = (const float*)d_in[14];
    const float* cls_b  = (const float*)d_in[15];
    float* out = (float*)d_out;

    // Workspace layout (floats):  tbuf | h | a | b | s   = 8 MB total
    float* ws   = (float*)d_ws;
    float* tbuf = ws;                                  // [B*N, HID]
    float* h    = tbuf + (size_t)B_ * N_ * HID_;       // [B*N, HID]
    float* aA   = h    + (size_t)B_ * N_ * HID_;       // [B*N, MSG]
    float* bA   = aA   + (size_t)B_ * N_ * MSG_;       // [B*N, MSG]
    float* sS   = bA   + (size_t)B_ * N_ * MSG_;       // [B*N, MSG]

    const int M = B_ * N_;                 // 2048 rows
    dim3 blk(128);
    dim3 gEnc(M / 64, HID_ / 16);          // 32 x 8
    dim3 gAB (M / 64, MSG_ / 16);          // 32 x 16

    // Encoder: tbuf = relu(X @ enc_w1^T + b1) ; h = tbuf @ enc_w2^T + b2
    wmma_gemm_f32<1><<<gEnc, blk, 0, stream>>>(X, DIN_, enc_w1, DIN_,
                                               enc_b1, 1.0f, nullptr,
                                               tbuf, HID_, DIN_);
    wmma_gemm_f32<2><<<gEnc, blk, 0, stream>>>(tbuf, HID_, enc_w2, HID_,
                                               enc_b2, 1.0f, nullptr,
                                               h, HID_, HID_);

    for (int l = 0; l < L_; ++l) {
        const float* w1l = msg_w1 + (size_t)l * MSG_ * (2 * HID_ + 1);
        const float* b1l = msg_b1 + (size_t)l * MSG_;
        const float* w2l = msg_w2 + (size_t)l * HID_ * MSG_;
        const float* b2l = msg_b2 + (size_t)l * HID_;

        // a = h @ wa^T ; b = h @ wb^T   (wa/wb are strided views of msg_w1[l])
        wmma_gemm_f32<0><<<gAB, blk, 0, stream>>>(h, HID_, w1l, 2 * HID_ + 1,
                                                  nullptr, 1.0f, nullptr,
                                                  aA, MSG_, HID_);
        wmma_gemm_f32<0><<<gAB, blk, 0, stream>>>(h, HID_, w1l + HID_, 2 * HID_ + 1,
                                                  nullptr, 1.0f, nullptr,
                                                  bA, MSG_, HID_);

        // s[b,n,m] = sum_j relu(a + b + W*ww + bias1)
        pairwise_msg<<<dim3(N_ / 16, B_), dim3(256), 0, stream>>>(
            aA, bA, W, w1l, b1l, sS);

        // h = h + relu(s @ msg_w2^T + N * msg_b2)
        wmma_gemm_f32<3><<<gEnc, blk, 0, stream>>>(sS, MSG_, w2l, MSG_,
                                                   b2l, (float)N_, h,
                                                   h, HID_, MSG_);
    }

    readout<<<dim3(B_), dim3(128), 0, stream>>>(h, ro_w1, ro_b1, ro_w2, ro_b2,
                                                cls_w, cls_b, out);
}